// PartialMultiheadAttention_29652454212057
// MI455X (gfx1250) — compile-verified
//
#include <hip/hip_runtime.h>

// ---------------------------------------------------------------------------
// PartialMultiheadAttention for MI455X (gfx1250, wave32, WMMA).
// Shapes: S=1024, B=8, E=1024, H=16, HD=64, BH=128. All f32.
// Memory-bound: ~1.1 GB of bias-read + prob-write => ~47us floor @ 23.3 TB/s.
// Math (69 GFLOP f32) done with V_WMMA_F32_16X16X4_F32 to keep full precision.
// ---------------------------------------------------------------------------

typedef __attribute__((ext_vector_type(2))) float v2f;
typedef __attribute__((ext_vector_type(8))) float v8f;

#define S_   1024
#define B_   8
#define E_   1024
#define H_   16
#define HD_  64
#define BH_  (B_ * H_)      // 128
#define N3E_ (3 * E_)       // 3072
#define SCALING_ 0.125f     // 64^-0.5

static __device__ __forceinline__ v8f wmma_f32_16x16x4(v2f a, v2f b, v8f c) {
  // 8 args: (neg_a, A, neg_b, B, c_mod, C, reuse_a, reuse_b)
  return __builtin_amdgcn_wmma_f32_16x16x4_f32(false, a, false, b, (short)0, c,
                                               false, false);
}

// ---------------------------------------------------------------------------
// Kernel 1: qkv = feats @ W^T + b, fused deswizzle into q/k/v planes.
// Each wave computes a 16x64 tile of the (8192 x 3072) result.
// A fragment (16x4 of feats) reused across 4 B tiles of W.
// Output scatter: f = h*192 + hd*3 + t  ->  plane t, [bh=b*16+h][s][hd].
// ---------------------------------------------------------------------------
__global__ __launch_bounds__(256) void qkv_gemm_kernel(
    const float* __restrict__ feats,   // (S,B,E) -> rows m = s*B + b
    const float* __restrict__ W,       // (3E, E) row-major
    const float* __restrict__ bqkv,    // (3E)
    float* __restrict__ qbuf,          // (BH,S,HD) ws
    float* __restrict__ kbuf,          // (BH,S,HD) ws
    float* __restrict__ vout)          // (BH,S,HD) tail of d_out
{
  const int lane = threadIdx.x & 31;
  const int wave = threadIdx.x >> 5;
  const int lo = lane & 15;
  const int hi = lane >> 4;

  const int ngroups = N3E_ / 64;                 // 48 N-groups of 64
  int job = blockIdx.x * 8 + wave;               // 3072 blocks * 8 waves = 24576
  int tm = job / ngroups;                        // 0..511  (M tile of 16)
  int tg = job % ngroups;                        // 0..47   (N group of 64)
  int m0 = tm * 16;
  int n0 = tg * 64;

  v8f acc[4] = {};
  // A row for this lane: M = lo for both lane halves (16x4 f32 layout).
  const float* arow = feats + (size_t)(m0 + lo) * E_;
  const float* brow0 = W + (size_t)(n0 + lo) * E_;

  for (int k = 0; k < E_; k += 4) {
    v2f a = *(const v2f*)(arow + k + 2 * hi);    // K = k+2*hi, k+2*hi+1
#pragma unroll
    for (int t = 0; t < 4; ++t) {
      // B[k..][n] = W[n][k..]; lanes 0-15 K=k,k+1, lanes 16-31 K=k+2,k+3
      v2f b = *(const v2f*)(brow0 + (size_t)t * 16 * E_ + k + 2 * hi);
      acc[t] = wmma_f32_16x16x4(a, b, acc[t]);
    }
  }

#pragma unroll
  for (int t = 0; t < 4; ++t) {
    int n = n0 + t * 16 + lo;                    // output feature (fixed/lane)
    float bias = bqkv[n];
    int h   = n / 192;
    int rem = n - h * 192;
    int hd  = rem / 3;
    int c3  = rem - hd * 3;                      // 0=q, 1=k, 2=v
#pragma unroll
    for (int r = 0; r < 8; ++r) {
      int m = m0 + r + 8 * hi;                   // C/D layout: M = r + 8*hi
      int s = m >> 3;                            // m = s*B + b, B=8
      int b = m & 7;
      int bh = b * H_ + h;
      size_t idx = ((size_t)bh * S_ + s) * HD_ + hd;
      float val = acc[t][r] + bias;
      if (c3 == 0)      qbuf[idx] = val;
      else if (c3 == 1) kbuf[idx] = val;
      else              vout[idx] = val;
    }
  }
}

// ---------------------------------------------------------------------------
// Kernel 2: scores = bias + 0.125 * q k^T, then row softmax.
// Block = (bh, 16-row strip). 8 waves x 8 column tiles cover 16x1024 scores,
// staged in 64KB LDS, then 2 rows/wave wave32 shuffle reductions.
// ---------------------------------------------------------------------------
__global__ __launch_bounds__(256) void attn_softmax_kernel(
    const float* __restrict__ qbuf,      // (BH,S,HD)
    const float* __restrict__ kbuf,      // (BH,S,HD)
    const float* __restrict__ attn_bias, // (BH,S,S)
    float* __restrict__ out)             // (BH,S,S)
{
  __shared__ float smem[16 * S_];        // 64 KB of the WGP's 320 KB LDS

  const int lane = threadIdx.x & 31;
  const int wave = threadIdx.x >> 5;
  const int lo = lane & 15;
  const int hi = lane >> 4;
  const int bh = blockIdx.x;
  const int row0 = blockIdx.y * 16;

  const float* qbase = qbuf + ((size_t)bh * S_ + row0 + lo) * HD_;
  const float* kbase = kbuf + ((size_t)bh * S_ + (size_t)(wave * 8) * 16 + lo) * HD_;

  v8f acc[8] = {};
  for (int k = 0; k < HD_; k += 4) {
    v2f a = *(const v2f*)(qbase + k + 2 * hi);   // q rows, reused across tiles
#pragma unroll
    for (int t = 0; t < 8; ++t) {
      // B = k^T : B[kk][n] = kmat[n][kk]
      v2f b = *(const v2f*)(kbase + (size_t)t * 16 * HD_ + k + 2 * hi);
      acc[t] = wmma_f32_16x16x4(a, b, acc[t]);
    }
  }

  // bias add + stage scores in LDS
#pragma unroll
  for (int t = 0; t < 8; ++t) {
    int col = (wave * 8 + t) * 16 + lo;
#pragma unroll
    for (int r = 0; r < 8; ++r) {
      int row = r + 8 * hi;
      float val = attn_bias[((size_t)bh * S_ + (row0 + row)) * S_ + col]
                + SCALING_ * acc[t][r];
      smem[row * S_ + col] = val;
    }
  }
  __syncthreads();

  // softmax: 2 rows per wave, wave32 shuffle reductions
#pragma unroll
  for (int rr = 0; rr < 2; ++rr) {
    int row = wave * 2 + rr;
    float* rowp = smem + row * S_;

    float mx = -3.402823466e38f;
    for (int c = lane; c < S_; c += 32) mx = fmaxf(mx, rowp[c]);
#pragma unroll
    for (int off = 16; off > 0; off >>= 1) mx = fmaxf(mx, __shfl_xor(mx, off, 32));

    float sum = 0.0f;
    for (int c = lane; c < S_; c += 32) {
      float e = __expf(rowp[c] - mx);
      rowp[c] = e;
      sum += e;
    }
#pragma unroll
    for (int off = 16; off > 0; off >>= 1) sum += __shfl_xor(sum, off, 32);
    float inv = 1.0f / sum;

    float* op = out + ((size_t)bh * S_ + (row0 + row)) * S_;
    for (int c = lane; c < S_; c += 32) op[c] = rowp[c] * inv;  // coalesced
  }
}

// ---------------------------------------------------------------------------
extern "C" void kernel_launch(void* const* d_in, const int* in_sizes, int n_in,
                              void* d_out, int out_size, void* d_ws,
                              size_t ws_size, hipStream_t stream) {
  const float* feats     = (const float*)d_in[0];  // (S,B,E)
  const float* attn_bias = (const float*)d_in[1];  // (B,H,S,S)
  const float* W_qkv     = (const float*)d_in[2];  // (3E,E)
  const float* b_qkv     = (const float*)d_in[3];  // (3E)

  float* out  = (float*)d_out;                       // (BH,S,S) probs
  float* vout = out + (size_t)BH_ * S_ * S_;         // (BH,S,HD) v (tail)

  float* qbuf = (float*)d_ws;                        // 33.5 MB
  float* kbuf = qbuf + (size_t)BH_ * S_ * HD_;       // 33.5 MB

  // 512 M-tiles * 48 N-groups = 24576 wave jobs / 8 waves per block
  qkv_gemm_kernel<<<dim3(3072), dim3(256), 0, stream>>>(
      feats, W_qkv, b_qkv, qbuf, kbuf, vout);

  attn_softmax_kernel<<<dim3(BH_, S_ / 16), dim3(256), 0, stream>>>(
      qbuf, kbuf, attn_bias, out);
}